// CombinedLoss_64922725646762
// MI455X (gfx1250) — compile-verified
//
#include <hip/hip_runtime.h>
#include <stdint.h>

// ---------------------------------------------------------------------------
// CombinedLoss on 256^3 f32 volume for gfx1250 (MI455X).
// Bandwidth-bound. Pass A streams yp/yt/x (192MB, x loaded non-temporal so
// L2 (192MB) retains yp/yt); pass C re-reads yp/yt (b128 loads, mostly L2
// hits).  15^3 box dilation = bit-packed separable OR-dilation (2MB mask,
// L2-resident).  Final partial-sum reduction runs on the matrix pipe via
// V_WMMA_F32_16X16X4_F32 with a ones B-matrix.
// ---------------------------------------------------------------------------

typedef float v2f __attribute__((ext_vector_type(2)));
typedef float v8f __attribute__((ext_vector_type(8)));

#define NTOT   (1u << 24)     // 256^3 elements
#define ROWS   65536          // 256*256 rows of 256 elements (W innermost)
#define WORDS  (NTOT / 32u)   // 524288 packed mask words
#define WPR    8              // mask words per row

#define PASSA_BLOCKS 2048
#define PASSC_BLOCKS 4096
#define NPART (PASSA_BLOCKS * 2 + PASSC_BLOCKS)   // 8192, multiple of 64

__device__ __forceinline__ float waveReduceAdd(float v) {
    v += __shfl_xor(v, 1, 32);
    v += __shfl_xor(v, 2, 32);
    v += __shfl_xor(v, 4, 32);
    v += __shfl_xor(v, 8, 32);
    v += __shfl_xor(v, 16, 32);
    return v;
}

// Pass A: stream yp/yt/x, accumulate sum(d^2) and sum(bone*d^2) per block,
// build hole mask bits, W-dilate (radius 7) each 256-bit row in registers,
// store packed mask.  One wave32 per row per iteration; per-lane b32 loads
// keep lane L == bit L so ballot_w32 packs the row directly.
__global__ void __launch_bounds__(256) passA_kernel(
    const float* __restrict__ yp, const float* __restrict__ yt,
    const float* __restrict__ x, uint32_t* __restrict__ mask,
    float* __restrict__ mseP, float* __restrict__ boneP) {
    const int lane = threadIdx.x & 31;
    const int wave = threadIdx.x >> 5;
    const int wgid = blockIdx.x * 8 + wave;
    const int nwaves = gridDim.x * 8;

    float s_mse = 0.0f, s_bone = 0.0f;

    for (int row = wgid; row < ROWS; row += nwaves) {
        const int base = row << 8;
        uint32_t w[8];
#pragma unroll
        for (int it = 0; it < 8; ++it) {
            const int e = base + (it << 5) + lane;   // coalesced 128B / wave
            const float a = yp[e];
            const float b = yt[e];
            // x is read exactly once in the whole loss -> non-temporal, so
            // the 192MB L2 keeps yp/yt resident for pass C.
            const float c = __builtin_nontemporal_load(x + e);
            const float d = a - b;
            const float d2 = d * d;
            const bool y1 = (b >= 0.5f);
            const bool x1 = (c >= 0.5f);
            s_mse += d2;
            s_bone += (y1 | x1) ? d2 : 0.0f;
            // hole = (y1 - x1) >= 0.5  <=>  y1 && !x1
            w[it] = __builtin_amdgcn_ballot_w32(y1 && !x1);
        }
        // OR-dilation composes: radius 7 = shifts by 1, 2, 4 (each both ways)
#pragma unroll
        for (int s = 1; s <= 4; s <<= 1) {
            uint32_t t[8];
#pragma unroll
            for (int j = 0; j < 8; ++j) {
                uint32_t lo = (w[j] << s) | (j > 0 ? (w[j - 1] >> (32 - s)) : 0u);
                uint32_t hi = (w[j] >> s) | (j < 7 ? (w[j + 1] << (32 - s)) : 0u);
                t[j] = w[j] | lo | hi;
            }
#pragma unroll
            for (int j = 0; j < 8; ++j) w[j] = t[j];
        }
        if (lane == 0) {   // ballot results are wave-uniform
            uint4* mp = (uint4*)(mask + (size_t)row * WPR);
            mp[0] = make_uint4(w[0], w[1], w[2], w[3]);
            mp[1] = make_uint4(w[4], w[5], w[6], w[7]);
        }
    }

    __shared__ float smse[8], sbone[8];
    const float rm = waveReduceAdd(s_mse);
    const float rb = waveReduceAdd(s_bone);
    if (lane == 0) { smse[wave] = rm; sbone[wave] = rb; }
    __syncthreads();
    if (threadIdx.x == 0) {
        float tm = 0.0f, tb = 0.0f;
        for (int i = 0; i < 8; ++i) { tm += smse[i]; tb += sbone[i]; }
        mseP[blockIdx.x] = tm;   // fixed slot per block -> deterministic
        boneP[blockIdx.x] = tb;
    }
}

// Bit-packed OR-dilation (radius 7) along H (axis=0) or D (axis=1).
// Word layout: t = d*2048 + h*8 + c ; bit b of word c = element w = c*32+b.
// Entirely L2-resident (2MB in, 2MB out, <=15 reads per word).
__global__ void __launch_bounds__(256) dilate_kernel(
    const uint32_t* __restrict__ in, uint32_t* __restrict__ out, int axis) {
    const int t = blockIdx.x * 256 + threadIdx.x;   // grid == WORDS/256
    const int c = t & 7;
    const int h = (t >> 3) & 255;
    const int d = t >> 11;
    const int pos    = axis ? d : h;
    const int stride = axis ? 2048 : 8;
    const int base   = axis ? ((h << 3) + c) : ((d << 11) + c);
    int k0 = pos - 7; if (k0 < 0) k0 = 0;
    int k1 = pos + 7; if (k1 > 255) k1 = 255;
    uint32_t acc = 0u;
    for (int k = k0; k <= k1; ++k) acc |= in[base + k * stride];
    out[t] = acc;
}

// Pass C: re-stream yp/yt (float4 / b128, non-temporal last-use; mostly L2
// hits thanks to pass A's cache-policy split), gate d^2 by the dilated mask.
__global__ void __launch_bounds__(256) passC_kernel(
    const float* __restrict__ yp, const float* __restrict__ yt,
    const uint32_t* __restrict__ mask, float* __restrict__ holeP) {
    const int lane = threadIdx.x & 31;
    const int wave = threadIdx.x >> 5;
    const int stride4 = gridDim.x * blockDim.x * 4;
    float s = 0.0f;
    typedef float f4 __attribute__((ext_vector_type(4)));
    for (int e = (blockIdx.x * 256 + threadIdx.x) * 4; e < (int)NTOT; e += stride4) {
        const f4 a = __builtin_nontemporal_load((const f4*)(yp + e));
        const f4 b = __builtin_nontemporal_load((const f4*)(yt + e));
        const uint32_t wbits = mask[e >> 5] >> (e & 31);  // 4 bits for this lane
#pragma unroll
        for (int j = 0; j < 4; ++j) {
            const float d = a[j] - b[j];
            s += ((wbits >> j) & 1u) ? d * d : 0.0f;
        }
    }
    __shared__ float sh[8];
    const float r = waveReduceAdd(s);
    if (lane == 0) sh[wave] = r;
    __syncthreads();
    if (threadIdx.x == 0) {
        float ts = 0.0f;
        for (int i = 0; i < 8; ++i) ts += sh[i];
        holeP[blockIdx.x] = ts;
    }
}

// Final reduction: one wave32 sums all partials on the matrix pipe.
// B = ones  =>  D[m][n] = sum_k A[m][k] + C[m][n]; accumulate C over chunks
// of 64 floats, then fold the 8 accumulator VGPRs + the lane16 half.
// Weights are all 1.0, so mse/bone/hole partials sum together; scale by 1/N.
__global__ void __launch_bounds__(32) final_kernel(
    const float* __restrict__ partials, float* __restrict__ out, int n) {
    const int lane = threadIdx.x;     // exactly one wave, EXEC all ones
    v8f c = {0.f, 0.f, 0.f, 0.f, 0.f, 0.f, 0.f, 0.f};
    v2f ones; ones.x = 1.0f; ones.y = 1.0f;
    for (int base = 0; base < n; base += 64) {
        v2f a;
        a.x = partials[base + lane];
        a.y = partials[base + 32 + lane];
        c = __builtin_amdgcn_wmma_f32_16x16x4_f32(
            /*neg_a=*/false, a, /*neg_b=*/false, ones,
            /*c_mod=*/(short)0, c, /*reuse_a=*/false, /*reuse_b=*/false);
    }
    float t = c[0] + c[1] + c[2] + c[3] + c[4] + c[5] + c[6] + c[7];
    t += __shfl_xor(t, 16, 32);       // combine rows 0-7 (lane0) + 8-15 (lane16)
    if (lane == 0) out[0] = t * (1.0f / 16777216.0f);
}

extern "C" void kernel_launch(void* const* d_in, const int* in_sizes, int n_in,
                              void* d_out, int out_size, void* d_ws, size_t ws_size,
                              hipStream_t stream) {
    const float* yp = (const float*)d_in[0];
    const float* yt = (const float*)d_in[1];
    const float* x  = (const float*)d_in[2];
    float* out = (float*)d_out;

    // Scratch layout: maskA(2MB) | maskB(2MB) | partials(32KB).  All scratch
    // that is read is written first each call (poison-safe, stateless).
    uint8_t* ws = (uint8_t*)d_ws;
    uint32_t* maskA = (uint32_t*)ws;
    uint32_t* maskB = (uint32_t*)(ws + (size_t)WORDS * 4u);
    float* partials = (float*)(ws + (size_t)WORDS * 8u);
    float* mseP  = partials;
    float* boneP = partials + PASSA_BLOCKS;
    float* holeP = partials + 2 * PASSA_BLOCKS;

    passA_kernel<<<PASSA_BLOCKS, 256, 0, stream>>>(yp, yt, x, maskA, mseP, boneP);
    dilate_kernel<<<WORDS / 256, 256, 0, stream>>>(maskA, maskB, 0); // H
    dilate_kernel<<<WORDS / 256, 256, 0, stream>>>(maskB, maskA, 1); // D
    passC_kernel<<<PASSC_BLOCKS, 256, 0, stream>>>(yp, yt, maskA, holeP);
    final_kernel<<<1, 32, 0, stream>>>(partials, out, NPART);
}